// TreeLSTM_34634616275576
// MI455X (gfx1250) — compile-verified
//
#include <hip/hip_runtime.h>
#include <math.h>

// ---------------------------------------------------------------------------
// TreeLSTM-ish single-vector forward for MI455X (gfx1250, wave32).
//
// Roofline: ~690 MB of f32 weights @ 23.3 TB/s => ~30us floor; bandwidth-bound.
// WMMA (V_WMMA_F32_16X16X4_F32) used where the math is a genuine GEMM:
//   phase 4:  W_u[j] (2048x2048) x c_leaf^T (2048x16)   -> N = 16 = L0
//   phase 5:  W_root (2048x2048) x s^T    (2048x8 pad16) -> N = 16 (8 zero cols)
// ---------------------------------------------------------------------------

#define D_DIM 4096
#define H_DIM 2048
#define L0_DIM 16
#define L1_DIM 8
#define TWO_SIGMA 10.0f

typedef float v2f __attribute__((ext_vector_type(2)));
typedef float v4f __attribute__((ext_vector_type(4)));
typedef float v8f __attribute__((ext_vector_type(8)));

// ------------------------- helpers -----------------------------------------

__device__ __forceinline__ float wave_reduce_add(float v) {
    // full 32-lane reduction (wave32)
    for (int off = 16; off >= 1; off >>= 1) v += __shfl_down(v, off, 32);
    return v;
}

// ------------------------- phase 1: RBF assign ------------------------------
// assign[l] = softmax_l( -||x - centers[l]||^2 / (2*sigma) ),  L0 = 16
__global__ void k_assign(const float* __restrict__ x,
                         const float* __restrict__ centers,
                         float* __restrict__ assign_out) {
    __shared__ float logits[L0_DIM];
    const int lane = threadIdx.x & 31;
    const int wid  = threadIdx.x >> 5;          // 16 waves, one per cluster

    const v4f* xc = (const v4f*)x;
    const v4f* cc = (const v4f*)(centers + (size_t)wid * D_DIM);
    float p = 0.f;
    for (int i = lane; i < D_DIM / 4; i += 32) {
        v4f xv = xc[i], cv = cc[i];
        float dx = xv.x - cv.x, dy = xv.y - cv.y, dz = xv.z - cv.z, dw = xv.w - cv.w;
        p += dx * dx + dy * dy + dz * dz + dw * dw;
    }
    p = wave_reduce_add(p);
    if (lane == 0) logits[wid] = -p / TWO_SIGMA;
    __syncthreads();

    if (threadIdx.x == 0) {
        float m = logits[0];
        for (int i = 1; i < L0_DIM; ++i) m = fmaxf(m, logits[i]);
        float sum = 0.f, e[L0_DIM];
        for (int i = 0; i < L0_DIM; ++i) { e[i] = __expf(logits[i] - m); sum += e[i]; }
        float inv = 1.0f / sum;
        for (int i = 0; i < L0_DIM; ++i) assign_out[i] = e[i] * inv;
    }
}

// ------------------------- phase 2: leaf GEMV -------------------------------
// c_leaf[l,h] = assign[l] * tanh( dot(W_leaf[l,h,:], x) + b_leaf[l,h] )
// 32768 rows x 4096; streams 537 MB -> THE bandwidth-critical kernel.
// wave-per-row, lane-strided 16B nontemporal loads (zero reuse of W_leaf).
__global__ void k_leaf(const float* __restrict__ x,
                       const float* __restrict__ W_leaf,
                       const float* __restrict__ b_leaf,
                       const float* __restrict__ assign_in,
                       float* __restrict__ c_leaf) {
    __shared__ float xs[D_DIM];                 // 16 KB, reused by 8 rows/block
    const int tid  = threadIdx.x;
    const int lane = tid & 31;
    const int wid  = tid >> 5;

    v4f* xs4 = (v4f*)xs;
    const v4f* xg4 = (const v4f*)x;
    for (int i = tid; i < D_DIM / 4; i += 256) xs4[i] = xg4[i];
    __syncthreads();

    const int row = blockIdx.x * 8 + wid;       // 0 .. 32767
    const v4f* wr = (const v4f*)(W_leaf + (size_t)row * D_DIM);
    float p = 0.f;
    for (int i = lane; i < D_DIM / 4; i += 32) {
        v4f w = __builtin_nontemporal_load(&wr[i]);   // TH=NT: don't pollute L2
        v4f xv = xs4[i];
        p += w.x * xv.x + w.y * xv.y + w.z * xv.z + w.w * xv.w;
    }
    p = wave_reduce_add(p);
    if (lane == 0) {
        const int l = row >> 11;                // row / H
        c_leaf[row] = assign_in[l] * tanhf(p + b_leaf[row]);
    }
}

// ------------------------- phase 3: input gates -----------------------------
// gate[l,j] = softmax_j( dot(c_leaf[l,:], W_i[j,:]) + b_i[j] ),  16 x 8
__global__ void k_gate(const float* __restrict__ c_leaf,
                       const float* __restrict__ W_i,
                       const float* __restrict__ b_i,
                       float* __restrict__ gate) {
    const int lane = threadIdx.x & 31;
    const int l    = threadIdx.x >> 5;          // 16 waves, one per leaf
    const v4f* cl = (const v4f*)(c_leaf + (size_t)l * H_DIM);

    float logit[L1_DIM];
    for (int j = 0; j < L1_DIM; ++j) {
        const v4f* wj = (const v4f*)(W_i + (size_t)j * H_DIM);
        float p = 0.f;
        for (int i = lane; i < H_DIM / 4; i += 32) {
            v4f c = cl[i], w = wj[i];
            p += c.x * w.x + c.y * w.y + c.z * w.z + c.w * w.w;
        }
        p = wave_reduce_add(p);
        if (lane == 0) logit[j] = p + b_i[j];
    }
    if (lane == 0) {
        float m = logit[0];
        for (int j = 1; j < L1_DIM; ++j) m = fmaxf(m, logit[j]);
        float sum = 0.f;
        for (int j = 0; j < L1_DIM; ++j) { logit[j] = __expf(logit[j] - m); sum += logit[j]; }
        float inv = 1.0f / sum;
        for (int j = 0; j < L1_DIM; ++j) gate[l * L1_DIM + j] = logit[j] * inv;
    }
}

// ------------------------- phase 4: WMMA GEMM + gated aggregation -----------
// For each j:  U = W_u[j] (MxK = 2048x2048) x B (KxN = 2048x16), B[k,n] = c_leaf[n,k].
// s[j,g] = sum_l gate[l,j] * tanh(U[g,l] + b_u[j,g]).
// One wave per (j, 16-row g-tile); K walked in 128-wide LDS slabs shared by the block.
#define SLAB 128
__global__ void k_u_wmma(const float* __restrict__ W_u,
                         const float* __restrict__ b_u,
                         const float* __restrict__ c_leaf,
                         const float* __restrict__ gate,
                         float* __restrict__ s_out) {
    __shared__ float Bs[L0_DIM * SLAB];         // 8 KB: c_leaf k-slab, layout [l][dk]
    const int tid  = threadIdx.x;
    const int lane = tid & 31;
    const int wid  = tid >> 5;
    const int wg   = blockIdx.x * 8 + wid;      // 0 .. 1023
    const int j    = wg >> 7;                   // / 128 g-tiles
    const int g0   = (wg & 127) * 16;

    const float* Wj = W_u + (size_t)j * H_DIM * H_DIM;
    const int r  = lane & 15;                   // A: row-in-tile / B,D: column N
    const int hi = lane >> 4;                   // K-pair select (ISA 16x4 f32 layout)

    v8f acc = {};                               // 16x16 f32 D tile
    for (int k0 = 0; k0 < H_DIM; k0 += SLAB) {
        for (int idx = tid; idx < L0_DIM * SLAB; idx += 256) {
            const int l = idx / SLAB, dk = idx % SLAB;
            Bs[idx] = c_leaf[(size_t)l * H_DIM + k0 + dk];
        }
        __syncthreads();
        for (int ks = 0; ks < SLAB; ks += 4) {
            const int kk = k0 + ks + 2 * hi;
            // A frag: lane r holds row g0+r, K pair {kk, kk+1}
            v2f a = *(const v2f*)&Wj[(size_t)(g0 + r) * H_DIM + kk];
            // B frag: lane r holds column n=r (leaf r), same K pair, from LDS
            v2f b = *(const v2f*)&Bs[r * SLAB + ks + 2 * hi];
            acc = __builtin_amdgcn_wmma_f32_16x16x4_f32(
                false, a, false, b, (short)0, acc, false, false);
        }
        __syncthreads();
    }

    // Epilogue: lane owns column n = r (leaf r), rows m = v + 8*hi.
    const float gj = gate[r * L1_DIM + j];
    for (int v = 0; v < 8; ++v) {
        const int m = v + 8 * hi;
        float val = gj * tanhf(acc[v] + b_u[(size_t)j * H_DIM + g0 + m]);
        for (int off = 8; off >= 1; off >>= 1) val += __shfl_xor(val, off, 32);
        if (r == 0) s_out[(size_t)j * H_DIM + g0 + m] = val;   // lanes 0 & 16
    }
}

// ------------------------- phase 5: root WMMA GEMM --------------------------
// out[h] = sum_{j<8} tanh( dot(s[j,:], W_root[h,:]) + b_root[h] )
// Same tiling; B = s^T padded to 16 columns with zeros (excluded in epilogue).
__global__ void k_root_wmma(const float* __restrict__ W_root,
                            const float* __restrict__ b_root,
                            const float* __restrict__ s_in,
                            float* __restrict__ out) {
    __shared__ float Bs[L0_DIM * SLAB];
    const int tid  = threadIdx.x;
    const int lane = tid & 31;
    const int wid  = tid >> 5;
    const int h0   = (blockIdx.x * 8 + wid) * 16;   // 128 h-tiles
    const int r  = lane & 15;
    const int hi = lane >> 4;

    v8f acc = {};
    for (int k0 = 0; k0 < H_DIM; k0 += SLAB) {
        for (int idx = tid; idx < L0_DIM * SLAB; idx += 256) {
            const int l = idx / SLAB, dk = idx % SLAB;
            Bs[idx] = (l < L1_DIM) ? s_in[(size_t)l * H_DIM + k0 + dk] : 0.0f;
        }
        __syncthreads();
        for (int ks = 0; ks < SLAB; ks += 4) {
            const int kk = k0 + ks + 2 * hi;
            v2f a = *(const v2f*)&W_root[(size_t)(h0 + r) * H_DIM + kk];
            v2f b = *(const v2f*)&Bs[r * SLAB + ks + 2 * hi];
            acc = __builtin_amdgcn_wmma_f32_16x16x4_f32(
                false, a, false, b, (short)0, acc, false, false);
        }
        __syncthreads();
    }

    for (int v = 0; v < 8; ++v) {
        const int m = v + 8 * hi;
        // column n=r is layer-1 node j=r; only j<8 are real
        float val = (r < L1_DIM) ? tanhf(acc[v] + b_root[h0 + m]) : 0.0f;
        for (int off = 8; off >= 1; off >>= 1) val += __shfl_xor(val, off, 32);
        if (r == 0) out[h0 + m] = val;
    }
}

// ------------------------- launcher -----------------------------------------

extern "C" void kernel_launch(void* const* d_in, const int* in_sizes, int n_in,
                              void* d_out, int out_size, void* d_ws, size_t ws_size,
                              hipStream_t stream) {
    const float* x       = (const float*)d_in[0];   // (4096)
    const float* centers = (const float*)d_in[1];   // (16,4096)
    const float* W_leaf  = (const float*)d_in[2];   // (16,2048,4096)
    const float* b_leaf  = (const float*)d_in[3];   // (16,2048)
    const float* W_i     = (const float*)d_in[4];   // (8,2048)
    const float* b_i     = (const float*)d_in[5];   // (8)
    const float* W_u     = (const float*)d_in[6];   // (8,2048,2048)
    const float* b_u     = (const float*)d_in[7];   // (8,2048)
    const float* W_root  = (const float*)d_in[8];   // (2048,2048)
    const float* b_root  = (const float*)d_in[9];   // (2048)
    float* out = (float*)d_out;                     // (1,2048)

    // workspace layout (floats)
    float* ws      = (float*)d_ws;
    float* assign  = ws;                            // 16
    float* c_leaf  = assign + L0_DIM;               // 16*2048
    float* gate    = c_leaf + L0_DIM * H_DIM;       // 16*8
    float* s_buf   = gate + L0_DIM * L1_DIM;        // 8*2048

    k_assign<<<dim3(1), dim3(512), 0, stream>>>(x, centers, assign);
    k_leaf<<<dim3((L0_DIM * H_DIM) / 8), dim3(256), 0, stream>>>(
        x, W_leaf, b_leaf, assign, c_leaf);
    k_gate<<<dim3(1), dim3(512), 0, stream>>>(c_leaf, W_i, b_i, gate);
    k_u_wmma<<<dim3(128), dim3(256), 0, stream>>>(W_u, b_u, c_leaf, gate, s_buf);
    k_root_wmma<<<dim3(16), dim3(256), 0, stream>>>(W_root, b_root, s_buf, out);
}